// GraphemeColourSynaesthesiaNet_22892175687790
// MI455X (gfx1250) — compile-verified
//
#include <hip/hip_runtime.h>
#include <hip/hip_bf16.h>
#include <math.h>

// ---------------------------------------------------------------------------
// GraphemeColourSynaesthesiaNet for MI455X (gfx1250), fp32 throughout.
//
// K is represented implicitly as K = -eta * (D*S^T - diag(sum_t d_t*s_t))
// (K0 == 0, updates are diag-masked rank-1), so K matvecs are low-rank.
// W @ x for the 8 distinct x vectors is precomputed with V_WMMA_F32_16X16X4_F32.
// ---------------------------------------------------------------------------

typedef float v2f __attribute__((ext_vector_type(2)));
typedef float v8f __attribute__((ext_vector_type(8)));

#define MDIM 2048
#define NDIM 1024
#define SDIM 8
#define XSTRIDE (NDIM + 4)   // LDS row stride: 4n mod 64 distinct -> no bank conflicts

// ---------------------------------------------------------------------------
// Phase 0: WX[k][i] = sum_j W[i][j] * X[k][j], k = 0..7 (padded to 16 cols).
// X staged in LDS (rows 8..15 zeroed) so the B operand is an unconditional
// ds_load_b64 — no EXEC-divergent guarded loads in the WMMA loop.
// 8 waves/block, one 16-row output tile per wave; fp32 WMMA, K-step = 4.
// A (16x4 f32): K = vgpr + 2*(lane>>4), M = lane&15.
// B (4x16 f32): K = vgpr + 2*(lane>>4), N = lane&15.
// D (16x16 f32): row = vgpr + 8*(lane>>4), col = lane&15.
// ---------------------------------------------------------------------------
__global__ __launch_bounds__(256) void wx_wmma_kernel(const float* __restrict__ W,
                                                      const float* __restrict__ X,
                                                      float* __restrict__ WX) {
    __shared__ float xsh[16 * XSTRIDE];
    for (int f = threadIdx.x; f < 16 * NDIM; f += 256) {
        const int r = f >> 10;          // / NDIM
        const int c = f & (NDIM - 1);   // % NDIM
        xsh[r * XSTRIDE + c] = (r < SDIM) ? X[(size_t)r * NDIM + c] : 0.0f;
    }
    __syncthreads();

    const int wave = threadIdx.x >> 5;
    const int lane = threadIdx.x & 31;
    const int half = lane >> 4;          // 0 or 1
    const int l15  = lane & 15;
    const int tile = blockIdx.x * 8 + wave;
    const int row  = tile * 16 + l15;    // A-matrix row for this lane

    v8f acc = {};
#pragma unroll 8
    for (int jb = 0; jb < NDIM; jb += 4) {
        const int k0 = jb + 2 * half;
        v2f a, b;
        a.x = W[(size_t)row * NDIM + k0 + 0];
        a.y = W[(size_t)row * NDIM + k0 + 1];
        b.x = xsh[l15 * XSTRIDE + k0 + 0];
        b.y = xsh[l15 * XSTRIDE + k0 + 1];
        acc = __builtin_amdgcn_wmma_f32_16x16x4_f32(
            /*neg_a=*/false, a, /*neg_b=*/false, b,
            /*c_mod=*/(short)0, acc, /*reuse_a=*/false, /*reuse_b=*/false);
    }
    // Scatter result: WX laid out [8][M] so per-step reads are coalesced over i.
    if (l15 < SDIM) {
#pragma unroll
        for (int vg = 0; vg < 8; ++vg) {
            const int r = tile * 16 + vg + 8 * half;
            WX[(size_t)l15 * MDIM + r] = acc[vg];
        }
    }
}

// ---------------------------------------------------------------------------
// Zero losses / diagvec / v_buf at the start of every launch (deterministic).
// ---------------------------------------------------------------------------
__global__ void init_kernel(float* __restrict__ losses, float* __restrict__ diagv,
                            float* __restrict__ vbuf, int steps) {
    const int i = blockIdx.x * blockDim.x + threadIdx.x;
    if (i < steps) losses[i] = 0.0f;
    if (i < MDIM) { diagv[i] = 0.0f; vbuf[i] = 0.0f; }
}

// ---------------------------------------------------------------------------
// q[c] = S[:,c] . v   (c = blockIdx.x, columns stored contiguously, M=2048)
// ---------------------------------------------------------------------------
__global__ __launch_bounds__(256) void proj_kernel(const float* __restrict__ Sm,
                                                   const float* __restrict__ v,
                                                   float* __restrict__ q) {
    __shared__ float red[256];
    const float* col = Sm + (size_t)blockIdx.x * MDIM;
    float acc = 0.0f;
    for (int i = threadIdx.x; i < MDIM; i += 256) acc += col[i] * v[i];
    red[threadIdx.x] = acc;
    __syncthreads();
    for (int s = 128; s > 0; s >>= 1) {
        if (threadIdx.x < s) red[threadIdx.x] += red[threadIdx.x + s];
        __syncthreads();
    }
    if (threadIdx.x == 0) q[blockIdx.x] = red[0];
}

// ---------------------------------------------------------------------------
// out[i] = sigmoid( wx[i] + K@v ) with K@v = -eta*(D q)[i] + eta*diagv[i]*v[i]
// ---------------------------------------------------------------------------
__global__ __launch_bounds__(256) void apply_kernel(const float* __restrict__ Dm,
                                                    const float* __restrict__ q,
                                                    const float* __restrict__ diagv,
                                                    const float* __restrict__ wx,
                                                    const float* __restrict__ v,
                                                    float* __restrict__ out,
                                                    int t, const float* __restrict__ eta_p) {
    const int i = blockIdx.x * blockDim.x + threadIdx.x;
    const float eta = eta_p[0];
    float acc = 0.0f;
    for (int tt = 0; tt < t; ++tt) acc += Dm[(size_t)tt * MDIM + i] * q[tt];
    const float u = wx[i] - eta * acc + eta * diagv[i] * v[i];
    out[i] = 1.0f / (1.0f + expf(-u));
}

// ---------------------------------------------------------------------------
// c_j = sum_i W[i][j]^2 * h_i   (h = (g(1-g))^2), invc_j = 1/c_j,
// loss += -0.5 * sum_j log(c_j).  64 columns per block, h staged in LDS.
// ---------------------------------------------------------------------------
__global__ __launch_bounds__(256) void col_kernel(const float* __restrict__ W,
                                                  const float* __restrict__ g,
                                                  float* __restrict__ invc,
                                                  float* __restrict__ loss) {
    __shared__ float hsh[MDIM];
    __shared__ float csh[4][64];
    __shared__ float lred[64];
    const int tid = threadIdx.x;
    for (int i = tid; i < MDIM; i += 256) {
        const float gg = g[i];
        const float gp = gg * (1.0f - gg);
        hsh[i] = gp * gp;
    }
    __syncthreads();
    const int j  = blockIdx.x * 64 + (tid & 63);
    const int rg = tid >> 6;
    float acc = 0.0f;
    for (int i = rg; i < MDIM; i += 4) {
        const float w = W[(size_t)i * NDIM + j];
        acc += w * w * hsh[i];
    }
    csh[rg][tid & 63] = acc;
    __syncthreads();
    if (rg == 0) {
        const float c = csh[0][tid] + csh[1][tid] + csh[2][tid] + csh[3][tid];
        invc[j]   = 1.0f / c;
        lred[tid] = logf(c);
    }
    __syncthreads();
    if (tid < 32) lred[tid] += lred[tid + 32];
    __syncthreads();
    if (tid < 16) lred[tid] += lred[tid + 16];
    __syncthreads();
    if (tid < 8)  lred[tid] += lred[tid + 8];
    __syncthreads();
    if (tid < 4)  lred[tid] += lred[tid + 4];
    __syncthreads();
    if (tid < 2)  lred[tid] += lred[tid + 2];
    __syncthreads();
    if (tid == 0) atomicAdd(loss, -0.5f * (lred[0] + lred[1]));
}

// ---------------------------------------------------------------------------
// Per row i: r_i = sum_j W[i][j]^2 * invc[j];  d_i = -h_i*(1-2g_i)*r_i.
// Append columns D[:,t]=d, S[:,t]=s; diagv += d*s; build next v = [colour(s2); s1].
// One wave per row (coalesced j), shuffle reduction.
// ---------------------------------------------------------------------------
__global__ __launch_bounds__(256) void row_kernel(const float* __restrict__ W,
                                                  const float* __restrict__ invc,
                                                  const float* __restrict__ g,
                                                  const float* __restrict__ s,
                                                  float* __restrict__ Dcol,
                                                  float* __restrict__ Scol,
                                                  float* __restrict__ diagv,
                                                  float* __restrict__ vnext) {
    const int lane = threadIdx.x & 31;
    const int wid  = threadIdx.x >> 5;
    const int i    = blockIdx.x * 8 + wid;
    const float* wr = W + (size_t)i * NDIM;
    float acc = 0.0f;
    for (int j = lane; j < NDIM; j += 32) {
        const float w = wr[j];
        acc += w * w * invc[j];
    }
    for (int off = 16; off > 0; off >>= 1) acc += __shfl_xor(acc, off, 32);
    if (lane == 0) {
        const float gg = g[i];
        const float gp = gg * (1.0f - gg);
        const float h  = gp * gp;
        const float d  = -h * (1.0f - 2.0f * gg) * acc;
        const float si = s[i];
        Dcol[i] = d;
        Scol[i] = si;
        diagv[i] += d * si;
        if (i < MDIM / 2) {
            // next v[0..1023] = colour_cat(s2) — snap to nearest of {0,.25,.5,.75}
            const float s2 = s[MDIM / 2 + i];
            float cc = floorf(s2 * 4.0f + 0.5f) * 0.25f;
            cc = fminf(fmaxf(cc, 0.0f), 0.75f);
            vnext[i] = cc;
        } else {
            // next v[1024..2047] = s1
            vnext[i] = s[i - MDIM / 2];
        }
    }
}

// ---------------------------------------------------------------------------
// Host orchestration
// ---------------------------------------------------------------------------
extern "C" void kernel_launch(void* const* d_in, const int* in_sizes, int n_in,
                              void* d_out, int out_size, void* d_ws, size_t ws_size,
                              hipStream_t stream) {
    (void)in_sizes; (void)n_in; (void)ws_size;
    const float* xs    = (const float*)d_in[0];   // (8, 1024) flat
    const float* W     = (const float*)d_in[1];   // (2048, 1024)
    /* d_in[2] = K0 (all zeros, unused: K kept in low-rank form) */
    const float* eta_p = (const float*)d_in[3];   // scalar eta_k
    /* d_in[4] = max_iter (unused by reference) */
    float* losses = (float*)d_out;                // (steps,) fp32

    const int steps = out_size;                   // 99
    const int div   = (steps + 1) / SDIM + 1;     // T//S + 1 = 13

    float* ws    = (float*)d_ws;
    float* WX    = ws;                                  // 8 * 2048
    float* Dm    = WX + (size_t)SDIM * MDIM;            // steps * 2048
    float* Sm    = Dm + (size_t)steps * MDIM;           // steps * 2048
    float* diagv = Sm + (size_t)steps * MDIM;           // 2048
    float* s_buf = diagv + MDIM;                        // 2048
    float* g_buf = s_buf + MDIM;                        // 2048
    float* v_buf = g_buf + MDIM;                        // 2048
    float* invc  = v_buf + MDIM;                        // 1024
    float* qb    = invc + NDIM;                         // steps (scratch)

    init_kernel<<<8, 256, 0, stream>>>(losses, diagv, v_buf, steps);
    wx_wmma_kernel<<<MDIM / (16 * 8), 256, 0, stream>>>(W, xs, WX);

    for (int tau = 0; tau < steps; ++tau) {
        const int k = (((tau + 1) / div) + (SDIM - 1)) % SDIM;  // (i//13 - 1) mod 8
        const float* wx = WX + (size_t)k * MDIM;

        // s = sigmoid(Wx + K @ [s2c; s1])
        if (tau > 0) proj_kernel<<<tau, 256, 0, stream>>>(Sm, v_buf, qb);
        apply_kernel<<<MDIM / 256, 256, 0, stream>>>(Dm, qb, diagv, wx, v_buf,
                                                     s_buf, tau, eta_p);
        // g = sigmoid(Wx + K @ s)
        if (tau > 0) proj_kernel<<<tau, 256, 0, stream>>>(Sm, s_buf, qb);
        apply_kernel<<<MDIM / 256, 256, 0, stream>>>(Dm, qb, diagv, wx, s_buf,
                                                     g_buf, tau, eta_p);
        // c, invc, E
        col_kernel<<<NDIM / 64, 256, 0, stream>>>(W, g_buf, invc, losses + tau);
        // r, d, append rank-1 factors, diag correction, next v (with colour_cat)
        row_kernel<<<MDIM / 8, 256, 0, stream>>>(W, invc, g_buf, s_buf,
                                                 Dm + (size_t)tau * MDIM,
                                                 Sm + (size_t)tau * MDIM,
                                                 diagv, v_buf);
    }
}